// Ortho_Model_35716948034062
// MI455X (gfx1250) — compile-verified
//
#include <hip/hip_runtime.h>
#include <math.h>

#ifndef __has_builtin
#define __has_builtin(x) 0
#endif

#define NT 256
#define CHUNK (NT * 4)   // floats per tile per array (4 KB)

typedef float v4f __attribute__((ext_vector_type(4)));
typedef int v4i_vs __attribute__((vector_size(16)));  // matches builtin param

// ---- CDNA5 async global->LDS path (guarded; falls back to plain loads) ----
#if __has_builtin(__builtin_amdgcn_global_load_async_to_lds_b128)
#define HAVE_ASYNC_LDS 1
#endif

#if defined(HAVE_ASYNC_LDS)
#if __has_builtin(__builtin_amdgcn_s_wait_asynccnt)
#define WAIT_ASYNC(n) __builtin_amdgcn_s_wait_asynccnt(n)
#else
#define WAIT_ASYNC(n) asm volatile("s_wait_asynccnt %0" ::"i"(n) : "memory")
#endif
#else
#define WAIT_ASYNC(n) ((void)0)
#endif

__device__ __forceinline__ void issue_tile16(const float* __restrict__ src,
                                             float* lds_dst) {
#if defined(HAVE_ASYNC_LDS)
  // async copy 16 bytes per lane: global (as1) -> LDS (as3), tracked by ASYNCcnt
  __builtin_amdgcn_global_load_async_to_lds_b128(
      (__attribute__((address_space(1))) v4i_vs*)(src),
      (__attribute__((address_space(3))) v4i_vs*)(lds_dst),
      0, 0);
#else
  *(v4f*)lds_dst = *(const v4f*)src;
#endif
}

// ---------------------------------------------------------------------------
// Stage 1: per-pair partial reductions of (w.w, w.g, g.g) via double-buffered
// async-to-LDS streaming. One launch per pair; gridDim.x = NB partial slots.
// part layout: [ww[NB] | wg[NB] | gg[NB]]
// ---------------------------------------------------------------------------
__global__ void __launch_bounds__(NT)
ortho_stage1(const float* __restrict__ w, const float* __restrict__ g,
             long long n, float* __restrict__ part) {
  __shared__ float tw[2][CHUNK];
  __shared__ float tg[2][CHUNK];
  __shared__ float red[3 * NT];

  const int tid = threadIdx.x;
  const long long nchunks = n / CHUNK;
  const long long c0 = blockIdx.x;
  const long long step = gridDim.x;

  v4f aww = {0.f, 0.f, 0.f, 0.f};
  v4f awg = {0.f, 0.f, 0.f, 0.f};
  v4f agg = {0.f, 0.f, 0.f, 0.f};

  int cur = 0;
  if (c0 < nchunks) {
    issue_tile16(w + c0 * CHUNK + tid * 4, &tw[0][tid * 4]);
    issue_tile16(g + c0 * CHUNK + tid * 4, &tg[0][tid * 4]);
  }
  for (long long c = c0; c < nchunks; c += step) {
    const long long cn = c + step;
    if (cn < nchunks) {
      issue_tile16(w + cn * CHUNK + tid * 4, &tw[cur ^ 1][tid * 4]);
      issue_tile16(g + cn * CHUNK + tid * 4, &tg[cur ^ 1][tid * 4]);
      WAIT_ASYNC(2);  // oldest tile (buf[cur]) complete; next tile in flight
    } else {
      WAIT_ASYNC(0);  // drain
    }
    __syncthreads();  // all waves' buf[cur] data visible in LDS
    v4f wv = *(const v4f*)&tw[cur][tid * 4];
    v4f gv = *(const v4f*)&tg[cur][tid * 4];
    aww += wv * wv;
    awg += wv * gv;
    agg += gv * gv;
    __syncthreads();  // buf[cur] free for the prefetch issued next iteration
    cur ^= 1;
  }

  float sww = aww.x + aww.y + aww.z + aww.w;
  float swg = awg.x + awg.y + awg.z + awg.w;
  float sgg = agg.x + agg.y + agg.z + agg.w;

  // scalar tail (sizes here are multiples of CHUNK, but stay generic)
  if (blockIdx.x == 0) {
    for (long long i = nchunks * CHUNK + tid; i < n; i += NT) {
      float wv = w[i], gv = g[i];
      sww = fmaf(wv, wv, sww);
      swg = fmaf(wv, gv, swg);
      sgg = fmaf(gv, gv, sgg);
    }
  }

  // deterministic block tree reduction
  red[tid] = sww;
  red[NT + tid] = swg;
  red[2 * NT + tid] = sgg;
  __syncthreads();
  for (int s = NT / 2; s > 0; s >>= 1) {
    if (tid < s) {
      red[tid] += red[tid + s];
      red[NT + tid] += red[NT + tid + s];
      red[2 * NT + tid] += red[2 * NT + tid + s];
    }
    __syncthreads();
  }
  if (tid == 0) {
    const long long nb = gridDim.x;
    part[blockIdx.x] = red[0];
    part[nb + blockIdx.x] = red[NT];
    part[2 * nb + blockIdx.x] = red[2 * NT];
  }
}

// ---------------------------------------------------------------------------
// Stage 2: one block per pair; finish reductions, compute proj & scale.
// scal[2p] = proj, scal[2p+1] = g_norm / (g_orth_norm + eps)
// ---------------------------------------------------------------------------
__global__ void __launch_bounds__(NT)
ortho_stage2(const float* __restrict__ part, float* __restrict__ scal, int nb) {
  __shared__ float red[3 * NT];
  const int tid = threadIdx.x;
  const float* base = part + (size_t)blockIdx.x * 3 * (size_t)nb;
  float a = 0.f, b = 0.f, c = 0.f;
  for (int j = tid; j < nb; j += NT) {
    a += base[j];
    b += base[nb + j];
    c += base[2 * nb + j];
  }
  red[tid] = a;
  red[NT + tid] = b;
  red[2 * NT + tid] = c;
  __syncthreads();
  for (int s = NT / 2; s > 0; s >>= 1) {
    if (tid < s) {
      red[tid] += red[tid + s];
      red[NT + tid] += red[NT + tid + s];
      red[2 * NT + tid] += red[2 * NT + tid + s];
    }
    __syncthreads();
  }
  if (tid == 0) {
    const float sww = red[0], swg = red[NT], sgg = red[2 * NT];
    const float proj = swg / (sww + 1e-30f);
    const float gn = sqrtf(sgg);
    float go2 = sgg - 2.0f * proj * swg + proj * proj * sww;  // ||g_orth||^2
    go2 = go2 > 0.f ? go2 : 0.f;
    const float scale = gn / (sqrtf(go2) + 1e-30f);
    scal[2 * blockIdx.x] = proj;
    scal[2 * blockIdx.x + 1] = scale;
  }
}

// ---------------------------------------------------------------------------
// Stage 3: out = (g - proj*w) * scale, vectorized, nontemporal stores.
// ---------------------------------------------------------------------------
__global__ void __launch_bounds__(NT)
ortho_stage3(const float* __restrict__ w, const float* __restrict__ g,
             float* __restrict__ out, long long n,
             const float* __restrict__ scal) {
  const float proj = scal[0];
  const float scale = scal[1];
  const long long nv = n / 4;
  const v4f* w4 = (const v4f*)w;
  const v4f* g4 = (const v4f*)g;
  v4f* o4 = (v4f*)out;
  const long long i0 = (long long)blockIdx.x * NT + threadIdx.x;
  const long long stride = (long long)gridDim.x * NT;
  for (long long i = i0; i < nv; i += stride) {
    v4f wv = w4[i];
    v4f gv = g4[i];
    v4f r = (gv - wv * proj) * scale;
    __builtin_nontemporal_store(r, o4 + i);
  }
  if (blockIdx.x == 0) {  // generic tail
    for (long long i = nv * 4 + threadIdx.x; i < n; i += NT)
      out[i] = (g[i] - proj * w[i]) * scale;
  }
}

// ---------------------------------------------------------------------------
extern "C" void kernel_launch(void* const* d_in, const int* in_sizes, int n_in,
                              void* d_out, int out_size, void* d_ws,
                              size_t ws_size, hipStream_t stream) {
  (void)n_in;
  (void)out_size;
  const float* w[4];
  const float* g[4];
  long long n[4];
  for (int p = 0; p < 4; ++p) {
    w[p] = (const float*)d_in[2 * p];
    g[p] = (const float*)d_in[2 * p + 1];
    n[p] = (long long)in_sizes[2 * p];
  }

  // workspace: 8 scalar floats + 4 pairs * 3 sums * NB partials
  int NB = 1024;
  while (NB > 1 && (size_t)(8 + 12 * NB) * sizeof(float) > ws_size) NB >>= 1;
  float* scal = (float*)d_ws;
  float* part = scal + 8;

  for (int p = 0; p < 4; ++p) {
    ortho_stage1<<<NB, NT, 0, stream>>>(w[p], g[p], n[p],
                                        part + (size_t)p * 3 * (size_t)NB);
  }
  ortho_stage2<<<4, NT, 0, stream>>>(part, scal, NB);

  float* out = (float*)d_out;
  long long off = 0;
  for (int p = 0; p < 4; ++p) {
    long long nv = n[p] / 4;
    long long blocks = (nv + NT - 1) / NT;
    if (blocks > 8192) blocks = 8192;
    if (blocks < 1) blocks = 1;
    ortho_stage3<<<(int)blocks, NT, 0, stream>>>(w[p], g[p], out + off, n[p],
                                                 scal + 2 * p);
    off += n[p];
  }
}